// CrazyAttentionLayer_670014898465
// MI455X (gfx1250) — compile-verified
//
#include <hip/hip_runtime.h>
#include <math.h>

typedef __attribute__((ext_vector_type(2))) float v2f;
typedef __attribute__((ext_vector_type(8))) float v8f;

#define NSAMP    16           // samples per block
#define NROWS    (NSAMP * 10) // token rows per block (inter stage)
#define NTHREADS 128          // 4 waves (wave32)

__launch_bounds__(NTHREADS)
__global__ void crazy_attn_fused(
    const float* __restrict__ state,
    const float* __restrict__ pos_intra,
    const float* __restrict__ wq_i, const float* __restrict__ bq_i,
    const float* __restrict__ wk_i, const float* __restrict__ bk_i,
    const float* __restrict__ wv_i, const float* __restrict__ bv_i,
    const float* __restrict__ pos_e,
    const float* __restrict__ wq_e, const float* __restrict__ bq_e,
    const float* __restrict__ wk_e, const float* __restrict__ bk_e,
    const float* __restrict__ wv_e, const float* __restrict__ bv_e,
    float* __restrict__ out, int B)
{
    // LDS staging (~62 KB total)
    __shared__ __align__(16) float s_x[NROWS * 20];     // inter input (x + pos_inter)
    __shared__ __align__(16) float s_w[3 * 20 * 32];    // Wq/Wk/Wv zero-padded N=21->32
    __shared__ float s_bias[3 * 32];                    // biases padded
    __shared__ float s_pose[200];                       // pos_inter (10,20)
    __shared__ float s_posi[10];                        // pos_intra (10,1)
    __shared__ float s_ip[12];                          // intra wq,bq,wk,bk,wv,bv (2 each)
    __shared__ float s_qkv[3][NROWS * 21];              // inter q,k,v

    const int tid         = threadIdx.x;
    const int sample_base = blockIdx.x * NSAMP;
    int nsamp = B - sample_base; if (nsamp > NSAMP) nsamp = NSAMP;
    const int nrows = nsamp * 10;

    // Prefetch this block's state region (global_prefetch_b8, 128B lines)
    {
        const char* p = (const char*)(state + (size_t)sample_base * 100);
        int nlines = (nsamp * 400 + 127) >> 7;
        if (tid < nlines) __builtin_prefetch((const void*)(p + tid * 128), 0, 0);
    }

    // ---- Phase 0: stage weights / pos into LDS -----------------------------
    for (int i = tid; i < 3 * 20 * 32; i += NTHREADS) {
        int m = i / 640, r = i - m * 640, k = r >> 5, n = r & 31;
        const float* w = (m == 0) ? wq_e : ((m == 1) ? wk_e : wv_e);
        s_w[i] = (n < 21) ? w[k * 21 + n] : 0.0f;
    }
    for (int i = tid; i < 3 * 32; i += NTHREADS) {
        int m = i >> 5, n = i & 31;
        const float* b = (m == 0) ? bq_e : ((m == 1) ? bk_e : bv_e);
        s_bias[i] = (n < 21) ? b[n] : 0.0f;
    }
    for (int i = tid; i < 200; i += NTHREADS) s_pose[i] = pos_e[i];
    if (tid < 10) s_posi[tid] = pos_intra[tid];
    if (tid < 2) {
        s_ip[0 + tid] = wq_i[tid]; s_ip[2 + tid]  = bq_i[tid];
        s_ip[4 + tid] = wk_i[tid]; s_ip[6 + tid]  = bk_i[tid];
        s_ip[8 + tid] = wv_i[tid]; s_ip[10 + tid] = bv_i[tid];
    }
    __syncthreads();

    // ---- Phase A: intra attention (Din=1, hd=1, 2 heads), one frame/thread -
    for (int f = tid; f < nrows; f += NTHREADS) {
        int sl = f / 10, fr = f - sl * 10;
        const float* xs = state + (size_t)(sample_base + sl) * 100 + fr * 10;
        float xp[10];
#pragma unroll
        for (int j = 0; j < 10; ++j) xp[j] = xs[j] + s_posi[j];
        float* xrow = &s_x[f * 20];
#pragma unroll
        for (int h = 0; h < 2; ++h) {
            const float wq = s_ip[h],     bq = s_ip[2 + h];
            const float wk = s_ip[4 + h], bk = s_ip[6 + h];
            const float wv = s_ip[8 + h], bv = s_ip[10 + h];
            float q[10], k[10], v[10];
#pragma unroll
            for (int j = 0; j < 10; ++j) {
                q[j] = fmaf(xp[j], wq, bq);
                k[j] = fmaf(xp[j], wk, bk);
                v[j] = fmaf(xp[j], wv, bv);
            }
#pragma unroll
            for (int s = 0; s < 10; ++s) {
                float mx = -3.4e38f;                       // hd=1 -> scale = 1
#pragma unroll
                for (int t = 0; t < 10; ++t) mx = fmaxf(mx, q[s] * k[t]);
                float sum = 0.0f, acc = 0.0f;
#pragma unroll
                for (int t = 0; t < 10; ++t) {
                    float e = __expf(q[s] * k[t] - mx);
                    sum += e; acc = fmaf(e, v[t], acc);
                }
                // intra (B*10,10,2) -> (B,10,20): row=frame, col=2*token+head
                xrow[2 * s + h] = acc / sum + s_pose[fr * 20 + 2 * s + h];
            }
        }
    }
    __syncthreads();

    // ---- Phase B: inter QKV projection via V_WMMA_F32_16X16X4_F32 ----------
    // M-tile = 16 token rows (sample-boundary-free, rows independent),
    // K = 20 as 5 steps of 4, N = 21 as two 16-wide tiles (padded).
    {
        const int wave = tid >> 5, lane = tid & 31;
        const int lrow = lane & 15, lhi = lane >> 4;
        const int ntiles = (nrows + 15) >> 4;
        for (int tile = wave; tile < ntiles; tile += NTHREADS / 32) {
            v2f a[5];
#pragma unroll
            for (int ks = 0; ks < 5; ++ks) {
                // A 16x4 f32 layout: lane l -> row l%16, K = kbase + 2*(l/16) + vgpr
                const float* p = &s_x[(tile * 16 + lrow) * 20 + 4 * ks + 2 * lhi];
                a[ks] = *(const v2f*)p;                    // 8B-aligned ds_load_b64
            }
#pragma unroll
            for (int m = 0; m < 3; ++m) {
#pragma unroll
                for (int nt = 0; nt < 2; ++nt) {
                    v8f acc = {};
#pragma unroll
                    for (int ks = 0; ks < 5; ++ks) {
                        const int kb = 4 * ks + 2 * lhi;
                        v2f b;                              // B 4x16: lane l -> col l%16
                        b.x = s_w[m * 640 + (kb + 0) * 32 + nt * 16 + lrow];
                        b.y = s_w[m * 640 + (kb + 1) * 32 + nt * 16 + lrow];
                        acc = __builtin_amdgcn_wmma_f32_16x16x4_f32(
                            false, a[ks], false, b, (short)0, acc, false, false);
                    }
                    const int col = nt * 16 + lrow;
                    const float bias = s_bias[m * 32 + col];
                    if (col < 21) {
#pragma unroll
                        for (int vv = 0; vv < 8; ++vv) {
                            // D layout: lane l, vgpr v -> row 8*(l/16)+v, col l%16
                            int row = tile * 16 + lhi * 8 + vv;
                            s_qkv[m][row * 21 + col] = acc[vv] + bias;
                        }
                    }
                }
            }
        }
    }
    __syncthreads();

    // ---- Phase C: inter attention (hd=21), one query row per thread --------
    const float rs21 = 0.21821789023599236f;               // 1/sqrt(21)
    for (int idx = tid; idx < nrows; idx += NTHREADS) {
        int sl = idx / 10, s = idx - sl * 10;
        float q[21];
        const float* qr = &s_qkv[0][idx * 21];
#pragma unroll
        for (int c = 0; c < 21; ++c) q[c] = qr[c];
        float sc[10]; float mx = -3.4e38f;
#pragma unroll
        for (int t = 0; t < 10; ++t) {
            const float* kr = &s_qkv[1][(sl * 10 + t) * 21];
            float d = 0.0f;
#pragma unroll
            for (int c = 0; c < 21; ++c) d = fmaf(q[c], kr[c], d);
            d *= rs21; sc[t] = d; mx = fmaxf(mx, d);
        }
        float sum = 0.0f;
#pragma unroll
        for (int t = 0; t < 10; ++t) { sc[t] = __expf(sc[t] - mx); sum += sc[t]; }
        const float inv = 1.0f / sum;
        float o[21];
#pragma unroll
        for (int c = 0; c < 21; ++c) o[c] = 0.0f;
#pragma unroll
        for (int t = 0; t < 10; ++t) {
            const float w = sc[t] * inv;
            const float* vr = &s_qkv[2][(sl * 10 + t) * 21];
#pragma unroll
            for (int c = 0; c < 21; ++c) o[c] = fmaf(w, vr[c], o[c]);
        }
        float* op = out + ((size_t)(sample_base + sl) * 10 + s) * 21;
#pragma unroll
        for (int c = 0; c < 21; ++c) op[c] = o[c];
    }
}

extern "C" void kernel_launch(void* const* d_in, const int* in_sizes, int n_in,
                              void* d_out, int out_size, void* d_ws, size_t ws_size,
                              hipStream_t stream) {
    const float* state     = (const float*)d_in[0];
    const float* pos_intra = (const float*)d_in[1];
    const float* wq_i      = (const float*)d_in[2];
    const float* bq_i      = (const float*)d_in[3];
    const float* wk_i      = (const float*)d_in[4];
    const float* bk_i      = (const float*)d_in[5];
    const float* wv_i      = (const float*)d_in[6];
    const float* bv_i      = (const float*)d_in[7];
    const float* pos_e     = (const float*)d_in[8];
    const float* wq_e      = (const float*)d_in[9];
    const float* bq_e      = (const float*)d_in[10];
    const float* wk_e      = (const float*)d_in[11];
    const float* bk_e      = (const float*)d_in[12];
    const float* wv_e      = (const float*)d_in[13];
    const float* bv_e      = (const float*)d_in[14];

    const int B = in_sizes[0] / 100;
    const int grid = (B + NSAMP - 1) / NSAMP;
    crazy_attn_fused<<<grid, NTHREADS, 0, stream>>>(
        state, pos_intra, wq_i, bq_i, wk_i, bk_i, wv_i, bv_i,
        pos_e, wq_e, bq_e, wk_e, bk_e, wv_e, bv_e,
        (float*)d_out, B);
}